// DiffusionMemory_18940805776178
// MI455X (gfx1250) — compile-verified
//
#include <hip/hip_runtime.h>
#include <math.h>

typedef __attribute__((ext_vector_type(2))) float v2f;
typedef __attribute__((ext_vector_type(8))) float v8f;
typedef __attribute__((ext_vector_type(4))) int   v4i;

#define NB    32
#define NH    16
#define ND    64
#define NSIZE 262144
#define NS    1024
#define TS    128
#define GAMMA_F 0.99f

#define AS1 __attribute__((address_space(1)))
#define AS3 __attribute__((address_space(3)))

// ---- async global->LDS b128 (ASYNCcnt path), per-lane gather ----
__device__ __forceinline__ void async_load_b128(const void* gptr, void* lptr) {
#if __has_builtin(__builtin_amdgcn_global_load_async_to_lds_b128)
  __builtin_amdgcn_global_load_async_to_lds_b128(
      (AS1 v4i*)(uintptr_t)gptr, (AS3 v4i*)(uintptr_t)lptr, 0, 0);
#else
  unsigned l = (unsigned)(uintptr_t)lptr;  // flat->LDS: low 32 bits are the DS address
  asm volatile("global_load_async_to_lds_b128 %0, %1, off"
               :: "v"(l), "v"(gptr) : "memory");
#endif
}
__device__ __forceinline__ void async_wait0() {
#if __has_builtin(__builtin_amdgcn_s_wait_asynccnt)
  __builtin_amdgcn_s_wait_asynccnt(0);
#else
  asm volatile("s_wait_asynccnt 0x0" ::: "memory");
#endif
}

// ---------------- phase A: logsumexp over heads ----------------
__global__ __launch_bounds__(256) void k_lse(const float* __restrict__ logit,
                                             float* __restrict__ lse) {
  int s = blockIdx.x * 256 + threadIdx.x;
  const float* row = logit + (size_t)s * NH;
  float m = row[0];
#pragma unroll
  for (int h = 1; h < NH; ++h) m = fmaxf(m, row[h]);
  float sum = 0.f;
#pragma unroll
  for (int h = 0; h < NH; ++h) sum += __expf(row[h] - m);
  lse[s] = m + __logf(sum);
}

// ---------------- deterministic gumbel perturbation ----------------
__device__ __forceinline__ unsigned hash_u32(unsigned x) {
  x ^= x >> 16; x *= 0x7FEB352Du;
  x ^= x >> 15; x *= 0x846CA68Bu;
  x ^= x >> 16;
  return x;
}
__device__ __forceinline__ float gumbel(unsigned bh, unsigned s) {
  unsigned r = hash_u32(s * 0x9E3779B9u + bh * 0x85EBCA6Bu + 0x243F6A88u);
  float u = (float)(r >> 8) * (1.0f / 16777216.0f);
  u = fminf(fmaxf(u, 1e-7f), 1.0f - 1e-7f);
  return -__logf(-__logf(u));
}

// ---------------- phase B: per-(b,h) top-S selection ----------------
__global__ __launch_bounds__(256) void k_select(const float* __restrict__ logit,
                                                const float* __restrict__ lse,
                                                int* __restrict__ idx_out) {
  __shared__ unsigned hist[2048];
  __shared__ unsigned s_thresh, s_above, s_hi, s_eq;
  const int bh  = blockIdx.x;
  const int h   = bh & (NH - 1);
  const int tid = threadIdx.x;

  for (int i = tid; i < 2048; i += 256) hist[i] = 0u;
  __syncthreads();

  for (int s = tid; s < NSIZE; s += 256) {
    float key = logit[(size_t)s * NH + h] - lse[s] + gumbel(bh, s);
    unsigned u = __float_as_uint(key);
    u = (u & 0x80000000u) ? ~u : (u | 0x80000000u);  // monotone ascending map
    atomicAdd(&hist[u >> 21], 1u);
  }
  __syncthreads();

  if (tid == 0) {
    unsigned cum = 0;
    int t = 2047;
    for (; t > 0; --t) {
      unsigned c = hist[t];
      if (cum + c >= NS) break;
      cum += c;
    }
    s_thresh = (unsigned)t; s_above = cum; s_hi = 0u; s_eq = 0u;
  }
  __syncthreads();
  const unsigned th = s_thresh, above = s_above;

  for (int s = tid; s < NSIZE; s += 256) {
    float key = logit[(size_t)s * NH + h] - lse[s] + gumbel(bh, s);
    unsigned u = __float_as_uint(key);
    u = (u & 0x80000000u) ? ~u : (u | 0x80000000u);
    unsigned b = u >> 21;
    if (b > th) {
      unsigned p = atomicAdd(&s_hi, 1u);
      if (p < NS) idx_out[(size_t)bh * NS + p] = s;
    } else if (b == th) {
      unsigned p = atomicAdd(&s_eq, 1u) + above;
      if (p < NS) idx_out[(size_t)bh * NS + p] = s;
    }
  }
}

// ---- gather one 128x64 f32 tile into LDS via async b128 (no VGPR staging) ----
__device__ __forceinline__ void gather_tile(float (*__restrict__ tile)[72],
                                            const float* __restrict__ src,
                                            const int* __restrict__ sidx,
                                            int h, int tid) {
  const int r    = tid >> 1;        // 128 rows, 2 threads per row
  const int half = (tid & 1) * 32;  // 32 floats (8 x b128) each
  const float* p = src + ((size_t)sidx[r] * NH + h) * ND + half;
  float* l = &tile[r][half];
#pragma unroll
  for (int i = 0; i < 8; ++i) async_load_b128(p + i * 4, l + i * 4);
}

// ---------------- phase C: attention + scatter, one block per (b,h) ----------------
__global__ __launch_bounds__(256) void k_attn(
    const float* __restrict__ qg,  const float* __restrict__ kst,
    const float* __restrict__ vst, const float* __restrict__ lst,
    const int* __restrict__ idx_g,
    float* __restrict__ out_lk, float* __restrict__ out_lv,
    float* __restrict__ out_k,  float* __restrict__ out_v,
    float* __restrict__ out_l) {
  __shared__ float qv[ND];
  __shared__ float scores[NS];
  __shared__ int   sidx[NS];
  __shared__ float tileK[TS][72];
  __shared__ float tileV[TS][72];
  __shared__ float red[16];
  __shared__ float lerp[2][ND];

  const int bh = blockIdx.x, b = bh >> 4, h = bh & 15;
  const int tid = threadIdx.x, wave = tid >> 5, lane = tid & 31;

  if (tid < ND) qv[tid] = qg[((size_t)b * NH + h) * ND + tid];
  for (int i = tid; i < NS; i += 256) sidx[i] = idx_g[(size_t)bh * NS + i];
  __syncthreads();

  const int m  = lane & 15;
  const int kk = (lane >> 4) << 1;  // K-pair select: 0 (lanes 0-15) / 2 (lanes 16-31)

  // ---- pass 1: scores[s] = (q . k_s) * D^-1/2 via v_wmma_f32_16x16x4_f32 ----
  for (int t0 = 0; t0 < NS; t0 += TS) {
    gather_tile(tileK, kst, &sidx[t0], h, tid);
    if (t0 + TS < NS && tid < TS)
      __builtin_prefetch(kst + ((size_t)sidx[t0 + TS + tid] * NH + h) * ND, 0, 0);
    async_wait0();
    __syncthreads();

    v8f c = {};
    const int srow = wave * 16 + m;  // A[m][k] = K[srow][d0+k]
#pragma unroll
    for (int d0 = 0; d0 < ND; d0 += 4) {
      v2f a, bq;
      a.x  = tileK[srow][d0 + kk];
      a.y  = tileK[srow][d0 + kk + 1];
      bq.x = qv[d0 + kk];            // B[k][n] = q[d0+k] broadcast over N
      bq.y = qv[d0 + kk + 1];
      c = __builtin_amdgcn_wmma_f32_16x16x4_f32(false, a, false, bq,
                                                (short)0, c, false, false);
    }
    if (lane == 0) {
#pragma unroll
      for (int j = 0; j < 8; ++j) scores[t0 + wave * 16 + j] = c[j] * 0.125f;
    } else if (lane == 16) {
#pragma unroll
      for (int j = 0; j < 8; ++j) scores[t0 + wave * 16 + 8 + j] = c[j] * 0.125f;
    }
    __syncthreads();
  }

  // ---- softmax over 1024 scores ----
  float mx = -3.4e38f;
  for (int i = tid; i < NS; i += 256) mx = fmaxf(mx, scores[i]);
  for (int o = 16; o > 0; o >>= 1) mx = fmaxf(mx, __shfl_xor(mx, o, 32));
  if (lane == 0) red[wave] = mx;
  __syncthreads();
  if (tid == 0) {
    float v = red[0];
    for (int w = 1; w < 8; ++w) v = fmaxf(v, red[w]);
    red[8] = v;
  }
  __syncthreads();
  mx = red[8];
  float ls = 0.f;
  for (int i = tid; i < NS; i += 256) {
    float e = __expf(scores[i] - mx);
    scores[i] = e;
    ls += e;
  }
  for (int o = 16; o > 0; o >>= 1) ls += __shfl_xor(ls, o, 32);
  if (lane == 0) red[wave] = ls;
  __syncthreads();
  if (tid == 0) {
    float v = 0.f;
    for (int w = 0; w < 8; ++w) v += red[w];
    red[9] = v;
  }
  __syncthreads();
  const float inv = 1.f / red[9];
  for (int i = tid; i < NS; i += 256) scores[i] *= inv;  // scores := attn
  __syncthreads();

  // ---- pass 2: lerp_k / lerp_v = attn-weighted row sums via WMMA ----
  const int isV = wave >> 2;        // waves 0-3 -> K, 4-7 -> V
  const int d0  = (wave & 3) * 16;  // M-tile of d
  v8f acc = {};
  for (int t0 = 0; t0 < NS; t0 += TS) {
    gather_tile(tileK, kst, &sidx[t0], h, tid);
    gather_tile(tileV, vst, &sidx[t0], h, tid);
    async_wait0();
    __syncthreads();
    float (*tile)[72] = isV ? tileV : tileK;
#pragma unroll 4
    for (int s0 = 0; s0 < TS; s0 += 4) {
      v2f a, bw;
      a.x  = tile[s0 + kk][d0 + m];      // A[m=d][k=s]
      a.y  = tile[s0 + kk + 1][d0 + m];
      bw.x = scores[t0 + s0 + kk];       // B[k][n] = attn[s] broadcast
      bw.y = scores[t0 + s0 + kk + 1];
      acc = __builtin_amdgcn_wmma_f32_16x16x4_f32(false, a, false, bw,
                                                  (short)0, acc, false, false);
    }
    __syncthreads();
  }
  if (lane == 0) {
#pragma unroll
    for (int j = 0; j < 8; ++j) lerp[isV][d0 + j] = acc[j];
  } else if (lane == 16) {
#pragma unroll
    for (int j = 0; j < 8; ++j) lerp[isV][d0 + 8 + j] = acc[j];
  }
  __syncthreads();

  if (tid < ND)
    __builtin_nontemporal_store(lerp[0][tid], &out_lk[((size_t)b * NH + h) * ND + tid]);
  else if (tid < 2 * ND)
    __builtin_nontemporal_store(lerp[1][tid - ND], &out_lv[((size_t)b * NH + h) * ND + (tid - ND)]);

  // ---- pass 3: scatter updated K/V rows and logits (non-temporal stores) ----
  for (int t0 = 0; t0 < NS; t0 += TS) {
    gather_tile(tileK, kst, &sidx[t0], h, tid);
    gather_tile(tileV, vst, &sidx[t0], h, tid);
    async_wait0();
    __syncthreads();
    for (int e = tid; e < TS * ND; e += 256) {
      int r = e >> 6, d = e & 63;
      int s = t0 + r;
      float a = scores[s];
      size_t off = ((size_t)sidx[s] * NH + h) * ND + d;
      __builtin_nontemporal_store((1.f - a) * tileK[r][d] + a * lerp[0][d], &out_k[off]);
      __builtin_nontemporal_store((1.f - a) * tileV[r][d] + a * lerp[1][d], &out_v[off]);
    }
    if (tid < TS) {
      int s = t0 + tid;
      int gi = sidx[s];
      __builtin_nontemporal_store(GAMMA_F * lst[(size_t)gi * NH + h] + scores[s],
                                  &out_l[(size_t)gi * NH + h]);
    }
    __syncthreads();
  }
}

// ---------------- host launcher ----------------
extern "C" void kernel_launch(void* const* d_in, const int* in_sizes, int n_in,
                              void* d_out, int out_size, void* d_ws, size_t ws_size,
                              hipStream_t stream) {
  (void)in_sizes; (void)n_in; (void)out_size; (void)ws_size;
  const float* q   = (const float*)d_in[0];
  const float* kst = (const float*)d_in[1];
  const float* vst = (const float*)d_in[2];
  const float* lst = (const float*)d_in[3];

  float* out    = (float*)d_out;
  float* out_lk = out;
  float* out_lv = out + (size_t)NB * NH * ND;
  float* out_k  = out_lv + (size_t)NB * NH * ND;
  float* out_v  = out_k + (size_t)NSIZE * NH * ND;
  float* out_l  = out_v + (size_t)NSIZE * NH * ND;

  float* lse = (float*)d_ws;
  int*   idx = (int*)((char*)d_ws + (size_t)NSIZE * sizeof(float));

  // passthrough copies of the stores (dominant traffic: ~4 GB -> memcpy path)
  (void)hipMemcpyAsync(out_k, kst, (size_t)NSIZE * NH * ND * sizeof(float),
                       hipMemcpyDeviceToDevice, stream);
  (void)hipMemcpyAsync(out_v, vst, (size_t)NSIZE * NH * ND * sizeof(float),
                       hipMemcpyDeviceToDevice, stream);
  (void)hipMemcpyAsync(out_l, lst, (size_t)NSIZE * NH * sizeof(float),
                       hipMemcpyDeviceToDevice, stream);

  k_lse<<<NSIZE / 256, 256, 0, stream>>>(lst, lse);
  k_select<<<NB * NH, 256, 0, stream>>>(lst, lse, idx);
  k_attn<<<NB * NH, 256, 0, stream>>>(q, kst, vst, lst, idx,
                                      out_lk, out_lv, out_k, out_v, out_l);
}